// Reduce_63359357551294
// MI455X (gfx1250) — compile-verified
//
#include <hip/hip_runtime.h>
#include <hip/hip_bf16.h>

typedef float v2f __attribute__((ext_vector_type(2)));
typedef float v8f __attribute__((ext_vector_type(8)));

#define NH        512
#define KTOT      1024          // NH * KERNEL_SIZE
#define LMAX      4096
#define BATCH     32
#define NSTEPS    12
#define XS_ROWS   32            // 32 columns staged per block
#define XS_STRIDE 268           // 256 + 12 pad: 8B-aligned rows, conflict-free wmma b64 reads
#define KCHUNK    256           // K elements per LDS stage (= 128 channel pairs)

// per-batch finishing step: first t where ceil(N / 2^(t+1)) <= 1
__global__ void step_of_kernel(const int* __restrict__ N, int* __restrict__ stepof) {
  int i = threadIdx.x;
  if (i < BATCH) {
    int cur = N[i];
    int s = NSTEPS - 1;
    for (int t = 0; t < NSTEPS; ++t) {
      cur = (cur + 1) >> 1;
      if (cur <= 1) { s = t; break; }
    }
    stepof[i] = s;
  }
}

// One tree-reduction step: masked conv(k=2,s=2) as WMMA GEMM + GRU-style gating.
// Block: 8 waves; wave w -> channel tile o0=(blockIdx.y*8+w)*16, TWO column tiles
// (32 cols) so each A (W) fetch feeds two WMMAs. x panel staged to LDS in 4
// K-chunks via the async-to-LDS engine (ASYNCcnt), masked columns via VALU path.
__global__ __launch_bounds__(256) void reduce_step_kernel(
    const float* __restrict__ hin,
    const float* __restrict__ W,
    const float* __restrict__ bias,
    const int*   __restrict__ Narr,
    const int*   __restrict__ stepof,
    float*       __restrict__ hout,
    float*       __restrict__ capbuf,
    int t)
{
  const int shin  = 12 - t;            // log2(L_in)
  const int Lin   = 1 << shin;
  const int shout = shin - 1;          // log2(L_out)
  const int Lout  = 1 << shout;

  __shared__ float xs[XS_ROWS * XS_STRIDE];

  const int tid     = threadIdx.x;
  const int lane    = tid & 31;
  const int laneLow = lane & 15;
  const int hi      = lane >> 4;       // wave32 half-select (WMMA operand layout)
  const int wave    = tid >> 5;
  const int o0      = (blockIdx.y * 8 + wave) * 16;

  const int n0 = blockIdx.x * 32;      // 32 output columns per block

  // this lane's two output columns (WMMA C/D: N = lane&15)
  const int c0 = n0 + laneLow,        c1 = c0 + 16;
  const int bb0 = c0 >> shout,        bb1 = c1 >> shout;
  const int pp0 = c0 & (Lout - 1),    pp1 = c1 & (Lout - 1);
  const int Nn0 = (Narr[bb0] + (2 << t) - 1) >> (t + 1);
  const int Nn1 = (Narr[bb1] + (2 << t) - 1) >> (t + 1);
  const bool needed = (pp0 < Nn0) || (pp1 < Nn1);
  if (__ballot(needed) == 0ull) return;   // tile fully past every batch's live length

  // loader mapping: lane -> one column, 8 i-groups sweep the chunk (coalesced 256B)
  const int kcol = tid & 31;
  const int ig   = tid >> 5;           // 0..7
  const int gcl  = n0 + kcol;
  const int bbl  = gcl >> shout;
  const int pl   = gcl & (Lout - 1);
  const int Ncur = (Narr[bbl] + (1 << t) - 1) >> t;   // mask: c < ceil(N/2^t)
  const float* srcCol = hin + (size_t)bbl * NH * Lin + 2 * pl;
  const bool fullValid = (2 * pl + 1) < Ncur;         // no masking on this column

  // LDS byte offset of this column's row (for async-to-LDS vdst)
  unsigned ldsColBase;
  {
    __attribute__((address_space(3))) float* p3 =
        (__attribute__((address_space(3))) float*)xs;
    ldsColBase = (unsigned)(size_t)p3 + (unsigned)(kcol * XS_STRIDE * 4);
  }

  v8f accL0 = {}, accL1 = {}, accR0 = {}, accR1 = {}, accG0 = {}, accG1 = {};

  // W[o,i,k] row-major, K index j = 2i+k: A lane<16 -> K {0,1}, lane>=16 -> K {2,3}
  const v2f*   Wbase      = (const v2f*)W + (size_t)(o0 + laneLow) * (KTOT / 2) + hi;
  const size_t gateStride = (size_t)NH * (KTOT / 2);

  for (int chunk = 0; chunk < 4; ++chunk) {
    const int iBase = chunk * (KCHUNK / 2);           // 128 channel pairs per chunk
    if (fullValid) {
      // pure copy -> async DMA into LDS, no VGPR return traffic
      unsigned lds = ldsColBase + (unsigned)(8 * ig);
      unsigned long long ga =
          (unsigned long long)(srcCol + (size_t)(iBase + ig) * Lin);
#pragma unroll
      for (int ii = 0; ii < 16; ++ii) {
        asm volatile("global_load_async_to_lds_b64 %0, %1, off"
                     :: "v"(lds), "v"(ga) : "memory");
        lds += 8u * 8u;                               // 8 i-steps * 2 floats
        ga  += (unsigned long long)(8 * Lin) * 4ull;
      }
    } else {
      // boundary column: load, zero-mask c >= ceil(N/2^t), store to LDS
#pragma unroll 4
      for (int ii = 0; ii < 16; ++ii) {
        const int iL = ig + 8 * ii;
        v2f v = *(const v2f*)(srcCol + (size_t)(iBase + iL) * Lin);
        if (2 * pl     >= Ncur) v.x = 0.f;
        if (2 * pl + 1 >= Ncur) v.y = 0.f;
        *(v2f*)&xs[kcol * XS_STRIDE + 2 * iL] = v;
      }
    }
    asm volatile("s_wait_asynccnt 0x0" ::: "memory");
    __syncthreads();

    const v2f*   Wp = Wbase + chunk * (KCHUNK / 4) * 2;   // chunk*128 v2f
    const float* x0 = &xs[laneLow * XS_STRIDE + 2 * hi];
    const float* x1 = &xs[(laneLow + 16) * XS_STRIDE + 2 * hi];
#pragma unroll 2
    for (int kk = 0; kk < KCHUNK / 4; ++kk) {
      v2f a0  = Wp[2 * kk];                    // l-gate rows (shared by both tiles)
      v2f a1  = Wp[2 * kk +     gateStride];   // r-gate rows
      v2f a2  = Wp[2 * kk + 2 * gateStride];   // g-gate rows
      v2f bx0 = *(const v2f*)(x0 + 4 * kk);
      v2f bx1 = *(const v2f*)(x1 + 4 * kk);
      accL0 = __builtin_amdgcn_wmma_f32_16x16x4_f32(false, a0, false, bx0, (short)0, accL0, false, false);
      accL1 = __builtin_amdgcn_wmma_f32_16x16x4_f32(false, a0, false, bx1, (short)0, accL1, false, false);
      accR0 = __builtin_amdgcn_wmma_f32_16x16x4_f32(false, a1, false, bx0, (short)0, accR0, false, false);
      accR1 = __builtin_amdgcn_wmma_f32_16x16x4_f32(false, a1, false, bx1, (short)0, accR1, false, false);
      accG0 = __builtin_amdgcn_wmma_f32_16x16x4_f32(false, a2, false, bx0, (short)0, accG0, false, false);
      accG1 = __builtin_amdgcn_wmma_f32_16x16x4_f32(false, a2, false, bx1, (short)0, accG1, false, false);
    }
    __syncthreads();
  }

  // ---- gating epilogue: C/D layout VGPR r -> M = r + hi*8, N = laneLow ----
  const int cap0 = (pp0 == 0) && (stepof[bb0] == t);
  const int cap1 = (pp1 == 0) && (stepof[bb1] == t);
#pragma unroll
  for (int r = 0; r < 8; ++r) {
    const int ch = o0 + r + hi * 8;
    const float bl = bias[ch], br = bias[NH + ch], bg = bias[2 * NH + ch];
    {
      float l  = accL0[r] + bl;
      float rr = tanhf(accR0[r] + br);
      float g  = 1.0f / (1.0f + __expf(-(accG0[r] + bg)));
      float val = l * g + rr * (1.0f - g);
      hout[((size_t)bb0 * NH + ch) * Lout + pp0] = val;
      if (cap0) capbuf[(size_t)bb0 * NH + ch] = val;
    }
    {
      float l  = accL1[r] + bl;
      float rr = tanhf(accR1[r] + br);
      float g  = 1.0f / (1.0f + __expf(-(accG1[r] + bg)));
      float val = l * g + rr * (1.0f - g);
      hout[((size_t)bb1 * NH + ch) * Lout + pp1] = val;
      if (cap1) capbuf[(size_t)bb1 * NH + ch] = val;
    }
  }
}

// out[rank(i)] = cap[i], rank by key = step_of*B + i (all keys distinct -> exact)
__global__ void order_kernel(const float* __restrict__ capbuf,
                             const int*   __restrict__ stepof,
                             float*       __restrict__ out) {
  const int i   = blockIdx.x;
  const int key = stepof[i] * BATCH + i;
  int rank = 0;
  for (int j = 0; j < BATCH; ++j)
    rank += ((stepof[j] * BATCH + j) < key) ? 1 : 0;
  for (int c = threadIdx.x; c < NH; c += blockDim.x)
    out[(size_t)rank * NH + c] = capbuf[(size_t)i * NH + c];
}

extern "C" void kernel_launch(void* const* d_in, const int* in_sizes, int n_in,
                              void* d_out, int out_size, void* d_ws, size_t ws_size,
                              hipStream_t stream) {
  const float* h = (const float*)d_in[0];
  const int*   N = (const int*)d_in[1];
  const float* W = (const float*)d_in[2];
  const float* b = (const float*)d_in[3];
  float* out = (float*)d_out;

  // workspace: ping (B*NH*2048 f32) | pong (B*NH*1024 f32) | capture (B*NH) | step_of (B)
  float* bufA = (float*)d_ws;
  float* bufB = bufA + (size_t)BATCH * NH * (LMAX / 2);
  float* cap  = bufB + (size_t)BATCH * NH * (LMAX / 4);
  int*   stp  = (int*)(cap + (size_t)BATCH * NH);

  step_of_kernel<<<1, 32, 0, stream>>>(N, stp);

  const float* cur = h;
  float* bufs[2] = {bufA, bufB};
  for (int t = 0; t < NSTEPS; ++t) {
    int Lout = LMAX >> (t + 1);
    int nBlk = (BATCH * Lout) / 32;          // exact: BATCH=32
    float* nxt = bufs[t & 1];
    reduce_step_kernel<<<dim3(nBlk, 4), 256, 0, stream>>>(cur, W, b, N, stp, nxt, cap, t);
    cur = nxt;
  }
  order_kernel<<<BATCH, 256, 0, stream>>>(cap, stp, out);
}